// GATv2_5858335391823
// MI455X (gfx1250) — compile-verified
//
#include <hip/hip_runtime.h>
#include <math.h>

typedef __attribute__((ext_vector_type(2))) float v2f;
typedef __attribute__((ext_vector_type(8))) float v8f;

#define HEADS 8
#define OUTC 16
#define HID 128
#define INC 64
#define NEG_SLOPE 0.2f

// float atomic max via signed/unsigned int ordering trick (init value must be -inf)
__device__ __forceinline__ void atomicMaxF(float* addr, float v) {
    if (v >= 0.0f) atomicMax((int*)addr, __float_as_int(v));
    else           atomicMin((unsigned int*)addr, (unsigned int)__float_as_int(v));
}

// ---------------------------------------------------------------------------
// K0: initialize accumulators (workspace is poisoned by harness; must re-init
//     every call).
// ---------------------------------------------------------------------------
__global__ void k_init(float* __restrict__ attn, float* __restrict__ segmax,
                       float* __restrict__ denom, float* __restrict__ pooled,
                       float* __restrict__ cnt, int nAttn, int nSeg, int nPool, int nG) {
    int i = blockIdx.x * blockDim.x + threadIdx.x;
    if (i < nAttn) attn[i] = 0.0f;
    if (i < nSeg) { segmax[i] = -INFINITY; denom[i] = 0.0f; }
    if (i < nPool) pooled[i] = 0.0f;
    if (i < nG)   cnt[i] = 0.0f;
}

// ---------------------------------------------------------------------------
// K1: fused node transforms  xl = x@Wl + bl, xr = x@Wr + br, xres = x@Wres
//     One block = 16 rows of x; 8 waves, wave w owns column tile w (16 cols).
//     fp32 WMMA 16x16x4, K = 64 -> 16 steps, 3 accumulators (48 WMMA/wave).
// ---------------------------------------------------------------------------
__global__ void __launch_bounds__(256)
k_node_gemm(const float* __restrict__ x,
            const float* __restrict__ Wl, const float* __restrict__ bl,
            const float* __restrict__ Wr, const float* __restrict__ br,
            const float* __restrict__ Wres,
            float* __restrict__ xl, float* __restrict__ xr,
            float* __restrict__ xres, int N) {
    const int lane = threadIdx.x & 31;
    const int wave = threadIdx.x >> 5;   // 0..7 -> column tile of 128 cols
    const int mloc = lane & 15;
    const int half = lane >> 4;          // 0 or 1
    const int row0 = blockIdx.x << 4;
    const int m    = row0 + mloc;        // A row this lane feeds
    const int n    = (wave << 4) + mloc; // global output column

    v8f accL = {0,0,0,0,0,0,0,0};
    v8f accR = {0,0,0,0,0,0,0,0};
    v8f accS = {0,0,0,0,0,0,0,0};

    #pragma unroll
    for (int kk = 0; kk < INC / 4; ++kk) {
        const int k = (kk << 2) + (half << 1);
        // A fragment: x[m][k], x[m][k+1] (k even -> 8B aligned)
        const float2 av = *(const float2*)(x + (size_t)m * INC + k);
        v2f a; a.x = av.x; a.y = av.y;
        // B fragments: W[k][n], W[k+1][n] for each of the 3 weight matrices
        const size_t w0 = (size_t)k * HID + n;
        v2f bL; bL.x = Wl[w0];   bL.y = Wl[w0 + HID];
        v2f bR; bR.x = Wr[w0];   bR.y = Wr[w0 + HID];
        v2f bS; bS.x = Wres[w0]; bS.y = Wres[w0 + HID];
        accL = __builtin_amdgcn_wmma_f32_16x16x4_f32(false, a, false, bL, (short)0, accL, false, false);
        accR = __builtin_amdgcn_wmma_f32_16x16x4_f32(false, a, false, bR, (short)0, accR, false, false);
        accS = __builtin_amdgcn_wmma_f32_16x16x4_f32(false, a, false, bS, (short)0, accS, false, false);
    }

    const float vbl = bl[n];
    const float vbr = br[n];
    #pragma unroll
    for (int r = 0; r < 8; ++r) {
        const int mm = row0 + r + (half << 3);  // D row: r (lanes 0-15) / r+8 (16-31)
        if (mm < N) {
            xl[(size_t)mm * HID + n]   = accL[r] + vbl;
            xr[(size_t)mm * HID + n]   = accR[r] + vbr;
            xres[(size_t)mm * HID + n] = accS[r];
        }
    }
}

// ---------------------------------------------------------------------------
// K2: per-(edge, head) attention logit + segment max.
//     edges e < E come from edge_index; e >= E are self loops.
// ---------------------------------------------------------------------------
__global__ void k_edge_alpha(const int* __restrict__ ei,
                             const float* __restrict__ xl, const float* __restrict__ xr,
                             const float* __restrict__ att,
                             float* __restrict__ alpha, float* __restrict__ segmax,
                             int E, int E2) {
    int idx = blockIdx.x * blockDim.x + threadIdx.x;
    if (idx >= E2 * HEADS) return;
    const int e = idx >> 3, h = idx & 7;
    int s, d;
    if (e < E) { s = ei[e]; d = ei[E + e]; } else { s = e - E; d = s; }
    const float* pl = xl + (size_t)s * HID + h * OUTC;
    const float* pr = xr + (size_t)d * HID + h * OUTC;
    const float* pa = att + h * OUTC;
    float acc = 0.0f;
    #pragma unroll
    for (int c = 0; c < OUTC; ++c) {
        float v = pl[c] + pr[c];
        v = v > 0.0f ? v : NEG_SLOPE * v;   // leaky_relu
        acc += pa[c] * v;
    }
    alpha[idx] = acc;
    atomicMaxF(&segmax[d * HEADS + h], acc);
}

// ---------------------------------------------------------------------------
// K3: alpha <- exp(alpha - segmax[dst]); denom[dst] += alpha
// ---------------------------------------------------------------------------
__global__ void k_edge_expsum(const int* __restrict__ ei,
                              float* __restrict__ alpha,
                              const float* __restrict__ segmax,
                              float* __restrict__ denom, int E, int E2) {
    int idx = blockIdx.x * blockDim.x + threadIdx.x;
    if (idx >= E2 * HEADS) return;
    const int e = idx >> 3, h = idx & 7;
    const int d = (e < E) ? ei[E + e] : (e - E);
    const float a = expf(alpha[idx] - segmax[d * HEADS + h]);
    alpha[idx] = a;
    atomicAdd(&denom[d * HEADS + h], a);
}

// ---------------------------------------------------------------------------
// K4: weighted scatter:  attn[dst][h][:] += (alpha/denom) * xl[src][h][:]
// ---------------------------------------------------------------------------
__global__ void k_edge_scatter(const int* __restrict__ ei,
                               const float* __restrict__ alpha,
                               const float* __restrict__ denom,
                               const float* __restrict__ xl,
                               float* __restrict__ attn, int E, int E2) {
    int idx = blockIdx.x * blockDim.x + threadIdx.x;
    if (idx >= E2 * HEADS) return;
    const int e = idx >> 3, h = idx & 7;
    int s, d;
    if (e < E) { s = ei[e]; d = ei[E + e]; } else { s = e - E; d = s; }
    const float w = alpha[idx] / denom[d * HEADS + h];
    const float* pl = xl + (size_t)s * HID + h * OUTC;
    float* po = attn + (size_t)d * HID + h * OUTC;
    #pragma unroll
    for (int c = 0; c < OUTC; ++c) atomicAdd(&po[c], w * pl[c]);
}

// ---------------------------------------------------------------------------
// K5: finalize: tmp = attn + xres + bias; y = ELU(tmp @ Wlin + blin);
//     pooled[batch[n]] += y; cnt[batch[n]] += 1.
//     One wave per 16-node tile, fp32 WMMA 16x16x4 over K=128 (32 steps).
// ---------------------------------------------------------------------------
__global__ void __launch_bounds__(128)
k_finalize(const float* __restrict__ attn, const float* __restrict__ xres,
           const float* __restrict__ bias,
           const float* __restrict__ Wlin, const float* __restrict__ blin,
           const int* __restrict__ batch,
           float* __restrict__ pooled, float* __restrict__ cnt,
           int N, int nTiles) {
    const int tile = blockIdx.x * (blockDim.x >> 5) + (threadIdx.x >> 5);
    if (tile >= nTiles) return;           // uniform per-wave exit (EXEC stays full)
    const int lane = threadIdx.x & 31;
    const int mloc = lane & 15;
    const int half = lane >> 4;
    const int nodeA = tile * 16 + mloc;   // node feeding A row mloc
    const int c = mloc;                   // output column this lane owns in B/D

    v8f acc = {0,0,0,0,0,0,0,0};
    #pragma unroll
    for (int kk = 0; kk < HID / 4; ++kk) {
        const int k = (kk << 2) + (half << 1);
        const size_t ro = (size_t)nodeA * HID + k;
        const float2 t0 = *(const float2*)(attn + ro);
        const float2 t1 = *(const float2*)(xres + ro);
        v2f a; a.x = t0.x + t1.x + bias[k];
               a.y = t0.y + t1.y + bias[k + 1];
        v2f b; b.x = Wlin[(size_t)k * OUTC + c];
               b.y = Wlin[(size_t)(k + 1) * OUTC + c];
        acc = __builtin_amdgcn_wmma_f32_16x16x4_f32(false, a, false, b, (short)0, acc, false, false);
    }

    const float vb = blin[c];
    #pragma unroll
    for (int r = 0; r < 8; ++r) {
        const int node = tile * 16 + r + (half << 3);
        if (node < N) {
            float y = acc[r] + vb;
            y = y > 0.0f ? y : (expf(y) - 1.0f);    // ELU
            const int g = batch[node];
            atomicAdd(&pooled[(size_t)g * OUTC + c], y);
            if (c == 0) atomicAdd(&cnt[g], 1.0f);   // lanes 0 & 16 cover all 16 rows
        }
    }
}

// ---------------------------------------------------------------------------
// K6: mean pool + 3-layer MLP head, one thread per graph (500 graphs).
// ---------------------------------------------------------------------------
__global__ void k_pool_mlp(const float* __restrict__ pooled, const float* __restrict__ cnt,
                           const float* __restrict__ W1, const float* __restrict__ b1,
                           const float* __restrict__ W2, const float* __restrict__ b2,
                           const float* __restrict__ W3, const float* __restrict__ b3,
                           float* __restrict__ out, int G) {
    int g = blockIdx.x * blockDim.x + threadIdx.x;
    if (g >= G) return;
    float c = cnt[g]; if (c < 1.0f) c = 1.0f;
    float s[16];
    #pragma unroll
    for (int j = 0; j < 16; ++j) s[j] = pooled[(size_t)g * 16 + j] / c;
    float h1[16];
    #pragma unroll
    for (int j = 0; j < 16; ++j) {
        float a = b1[j];
        for (int i = 0; i < 16; ++i) a += s[i] * W1[i * 16 + j];
        h1[j] = a > 0.0f ? a : 0.0f;
    }
    float h2[32];
    #pragma unroll
    for (int j = 0; j < 32; ++j) {
        float a = b2[j];
        for (int i = 0; i < 16; ++i) a += h1[i] * W2[i * 32 + j];
        h2[j] = a > 0.0f ? a : 0.0f;
    }
    #pragma unroll
    for (int j = 0; j < 5; ++j) {
        float a = b3[j];
        for (int i = 0; i < 32; ++i) a += h2[i] * W3[i * 5 + j];
        out[(size_t)g * 5 + j] = a;
    }
}

// ---------------------------------------------------------------------------
extern "C" void kernel_launch(void* const* d_in, const int* in_sizes, int n_in,
                              void* d_out, int out_size, void* d_ws, size_t ws_size,
                              hipStream_t stream) {
    const float* x    = (const float*)d_in[0];
    const int*   ei   = (const int*)d_in[1];
    const int*   batch= (const int*)d_in[2];
    const float* Wl   = (const float*)d_in[3];
    const float* bl   = (const float*)d_in[4];
    const float* Wr   = (const float*)d_in[5];
    const float* br   = (const float*)d_in[6];
    const float* att  = (const float*)d_in[7];
    const float* Wres = (const float*)d_in[8];
    const float* bias = (const float*)d_in[9];
    const float* Wlin = (const float*)d_in[10];
    const float* blin = (const float*)d_in[11];
    const float* W1   = (const float*)d_in[12];
    const float* b1   = (const float*)d_in[13];
    const float* W2   = (const float*)d_in[14];
    const float* b2   = (const float*)d_in[15];
    const float* W3   = (const float*)d_in[16];
    const float* b3   = (const float*)d_in[17];

    const int N  = in_sizes[0] / INC;     // 50000
    const int E  = in_sizes[1] / 2;       // 800000
    const int E2 = E + N;                 // + self loops
    const int G  = out_size / 5;          // 500

    // workspace layout (floats)
    float* ws = (float*)d_ws;
    float* xl     = ws; ws += (size_t)N * HID;
    float* xr     = ws; ws += (size_t)N * HID;
    float* xres   = ws; ws += (size_t)N * HID;
    float* attn   = ws; ws += (size_t)N * HID;
    float* alpha  = ws; ws += (size_t)E2 * HEADS;
    float* segmax = ws; ws += (size_t)N * HEADS;
    float* denom  = ws; ws += (size_t)N * HEADS;
    float* pooled = ws; ws += (size_t)G * OUTC;
    float* cnt    = ws; ws += G;

    float* outp = (float*)d_out;

    // K0: init accumulators
    {
        int total = N * HID;
        int blocks = (total + 255) / 256;
        k_init<<<blocks, 256, 0, stream>>>(attn, segmax, denom, pooled, cnt,
                                           N * HID, N * HEADS, G * OUTC, G);
    }
    // K1: node transforms (WMMA)
    {
        int blocks = (N + 15) / 16;
        k_node_gemm<<<blocks, 256, 0, stream>>>(x, Wl, bl, Wr, br, Wres, xl, xr, xres, N);
    }
    // K2/K3/K4: edge phase
    {
        int total = E2 * HEADS;
        int blocks = (total + 255) / 256;
        k_edge_alpha<<<blocks, 256, 0, stream>>>(ei, xl, xr, att, alpha, segmax, E, E2);
        k_edge_expsum<<<blocks, 256, 0, stream>>>(ei, alpha, segmax, denom, E, E2);
        k_edge_scatter<<<blocks, 256, 0, stream>>>(ei, alpha, denom, xl, attn, E, E2);
    }
    // K5: finalize + pool accumulate (WMMA)
    {
        int nTiles = (N + 15) / 16;
        int wavesPerBlock = 4;                     // 128 threads
        int blocks = (nTiles + wavesPerBlock - 1) / wavesPerBlock;
        k_finalize<<<blocks, 128, 0, stream>>>(attn, xres, bias, Wlin, blin, batch,
                                               pooled, cnt, N, nTiles);
    }
    // K6: mean pool + MLP head
    {
        int blocks = (G + 127) / 128;
        k_pool_mlp<<<blocks, 128, 0, stream>>>(pooled, cnt, W1, b1, W2, b2, W3, b3, outp, G);
    }
}